// HeteroGNNModel_61083024883722
// MI455X (gfx1250) — compile-verified
//
#include <hip/hip_runtime.h>

typedef __attribute__((ext_vector_type(2))) float v2f;
typedef __attribute__((ext_vector_type(8))) float v8f;

#define N_LOC   100000
#define N_EVT   100000
#define N_EDGE  1600000
#define F_LOC   128
#define F_EVT   128
#define QID_DIM 16
#define HID     64

// D = A(16x4, fp32) * B(4x16, fp32) + C   -> v_wmma_f32_16x16x4_f32
static __device__ inline v8f wmma4(v2f a, v2f b, v8f c) {
  return __builtin_amdgcn_wmma_f32_16x16x4_f32(
      /*neg_a=*/false, a, /*neg_b=*/false, b,
      /*c_mod=*/(short)0, c, /*reuse_a=*/false, /*reuse_b=*/false);
}

// ---------------------------------------------------------------------------
// zero-fill workspace region (agg + cnt)
// ---------------------------------------------------------------------------
__global__ void zero_f32(float* __restrict__ p, long long n) {
  long long i = (long long)blockIdx.x * blockDim.x + threadIdx.x;
  long long stride = (long long)gridDim.x * blockDim.x;
  for (; i < n; i += stride) p[i] = 0.0f;
}

// ---------------------------------------------------------------------------
// evt_x = relu(evt_feat @ W_evt + b_evt)      [N_EVT x 64]
// block = 128 threads (4 waves); each wave owns one 16x16 N-tile; 16 rows/blk
// ---------------------------------------------------------------------------
__global__ void evt_gemm(const float* __restrict__ A, const float* __restrict__ W,
                         const float* __restrict__ bias, float* __restrict__ out) {
  const int wave = threadIdx.x >> 5;
  const int lane = threadIdx.x & 31;
  const int hi = lane >> 4;      // which K half-pair this lane carries
  const int lo = lane & 15;      // row (A) / col (B,C)
  const int m0 = blockIdx.x * 16;
  const int n0 = wave * 16;
  const int row = m0 + lo;

  v8f c = {};
  const float* arow = A + (size_t)row * F_EVT + 2 * hi;
  const float* wcol = W + n0 + lo + 2 * hi * HID;
  for (int k0 = 0; k0 < F_EVT; k0 += 4) {
    v2f a, b;
    a.x = arow[k0];
    a.y = arow[k0 + 1];
    b.x = wcol[k0 * HID];
    b.y = wcol[(k0 + 1) * HID];
    c = wmma4(a, b, c);
  }
  const float bv = bias[n0 + lo];
#pragma unroll
  for (int r = 0; r < 8; ++r) {
    float v = c[r] + bv;
    out[(size_t)(m0 + r + 8 * hi) * HID + n0 + lo] = fmaxf(v, 0.0f);
  }
}

// ---------------------------------------------------------------------------
// loc_x = relu([loc_feat || emb[qid]] @ W_loc + b_loc)   K = 128 + 16 = 144
// ---------------------------------------------------------------------------
__global__ void loc_gemm(const float* __restrict__ A, const int* __restrict__ qid,
                         const float* __restrict__ emb, const float* __restrict__ W,
                         const float* __restrict__ bias, float* __restrict__ out) {
  const int wave = threadIdx.x >> 5;
  const int lane = threadIdx.x & 31;
  const int hi = lane >> 4;
  const int lo = lane & 15;
  const int m0 = blockIdx.x * 16;
  const int n0 = wave * 16;
  const int row = m0 + lo;

  v8f c = {};
  const float* arow = A + (size_t)row * F_LOC + 2 * hi;
  const float* wcol = W + n0 + lo + 2 * hi * HID;
  for (int k0 = 0; k0 < F_LOC; k0 += 4) {           // dense-feature part
    v2f a, b;
    a.x = arow[k0];
    a.y = arow[k0 + 1];
    b.x = wcol[k0 * HID];
    b.y = wcol[(k0 + 1) * HID];
    c = wmma4(a, b, c);
  }
  const int q = qid[row];                            // qid-embedding part
  const float* erow = emb + (size_t)q * QID_DIM + 2 * hi;
  const float* wcol2 = W + (size_t)F_LOC * HID + n0 + lo + 2 * hi * HID;
  for (int k0 = 0; k0 < QID_DIM; k0 += 4) {
    v2f a, b;
    a.x = erow[k0];
    a.y = erow[k0 + 1];
    b.x = wcol2[k0 * HID];
    b.y = wcol2[(k0 + 1) * HID];
    c = wmma4(a, b, c);
  }
  const float bv = bias[n0 + lo];
#pragma unroll
  for (int r = 0; r < 8; ++r) {
    float v = c[r] + bv;
    out[(size_t)(m0 + r + 8 * hi) * HID + n0 + lo] = fmaxf(v, 0.0f);
  }
}

// ---------------------------------------------------------------------------
// scatter: agg[dst] += evt_x[src]; cnt[dst] += 1   (L2-resident fp32 atomics)
// 16 lanes per edge, each lane handles 4 contiguous H channels (float4 load)
// ---------------------------------------------------------------------------
__global__ void scatter_edges(const int* __restrict__ edge,
                              const float* __restrict__ evt_x,
                              float* __restrict__ agg, float* __restrict__ cnt) {
  const long long t = (long long)blockIdx.x * blockDim.x + threadIdx.x;
  const int e = (int)(t >> 4);
  const int l = (int)(t & 15);
  if (e >= N_EDGE) return;
  const int src = edge[e];
  const int dst = edge[N_EDGE + e];
  const float4 v = *(const float4*)(evt_x + (size_t)src * HID + l * 4);
  float* p = agg + (size_t)dst * HID + l * 4;
  __hip_atomic_fetch_add(p + 0, v.x, __ATOMIC_RELAXED, __HIP_MEMORY_SCOPE_AGENT);
  __hip_atomic_fetch_add(p + 1, v.y, __ATOMIC_RELAXED, __HIP_MEMORY_SCOPE_AGENT);
  __hip_atomic_fetch_add(p + 2, v.z, __ATOMIC_RELAXED, __HIP_MEMORY_SCOPE_AGENT);
  __hip_atomic_fetch_add(p + 3, v.w, __ATOMIC_RELAXED, __HIP_MEMORY_SCOPE_AGENT);
  if (l == 0)
    __hip_atomic_fetch_add(cnt + dst, 1.0f, __ATOMIC_RELAXED, __HIP_MEMORY_SCOPE_AGENT);
}

// ---------------------------------------------------------------------------
// fused: mean = agg/max(cnt,1);  loc2 = relu(mean@W_l + b_l + loc_x@W_r);
//        h = relu(loc2@W_h1 + b_h1);  logits = h@W_h2 + b_h2
// block = 128 threads (4 waves) per 16 rows; loc2 and h staged in LDS
// ---------------------------------------------------------------------------
__global__ void sage_head(const float* __restrict__ agg, const float* __restrict__ cnt,
                          const float* __restrict__ loc_x,
                          const float* __restrict__ Wl, const float* __restrict__ bl,
                          const float* __restrict__ Wr,
                          const float* __restrict__ Wh1, const float* __restrict__ bh1,
                          const float* __restrict__ Wh2, const float* __restrict__ bh2,
                          float* __restrict__ out) {
  __shared__ float lds_x[16][HID];     // loc2 tile
  __shared__ float lds_h[16][HID / 2]; // h tile

  const int wave = threadIdx.x >> 5;
  const int lane = threadIdx.x & 31;
  const int hi = lane >> 4;
  const int lo = lane & 15;
  const int m0 = blockIdx.x * 16;
  const int row = m0 + lo;
  const float inv = 1.0f / fmaxf(cnt[row], 1.0f);

  // ---- stage 1: loc2 = relu(mean@Wl + bl + loc_x@Wr), this wave's 16 cols
  {
    const int n0 = wave * 16;
    v8f c = {};
    const float* am = agg + (size_t)row * HID + 2 * hi;
    const float* as = loc_x + (size_t)row * HID + 2 * hi;
    const float* wl = Wl + n0 + lo + 2 * hi * HID;
    const float* wr = Wr + n0 + lo + 2 * hi * HID;
    for (int k0 = 0; k0 < HID; k0 += 4) {
      v2f a, b;
      a.x = am[k0] * inv;  a.y = am[k0 + 1] * inv;
      b.x = wl[k0 * HID];  b.y = wl[(k0 + 1) * HID];
      c = wmma4(a, b, c);
    }
    for (int k0 = 0; k0 < HID; k0 += 4) {
      v2f a, b;
      a.x = as[k0];        a.y = as[k0 + 1];
      b.x = wr[k0 * HID];  b.y = wr[(k0 + 1) * HID];
      c = wmma4(a, b, c);
    }
    const float bv = bl[n0 + lo];
#pragma unroll
    for (int r = 0; r < 8; ++r)
      lds_x[r + 8 * hi][n0 + lo] = fmaxf(c[r] + bv, 0.0f);
  }
  __syncthreads();

  // ---- stage 2: h = relu(loc2 @ Wh1 + bh1), N = 32 -> waves 0,1
  if (wave < 2) {
    const int n0 = wave * 16;
    v8f c = {};
    const float* wc = Wh1 + n0 + lo + 2 * hi * (HID / 2);
    for (int k0 = 0; k0 < HID; k0 += 4) {
      v2f a, b;
      a.x = lds_x[lo][k0 + 2 * hi];
      a.y = lds_x[lo][k0 + 1 + 2 * hi];
      b.x = wc[k0 * (HID / 2)];
      b.y = wc[(k0 + 1) * (HID / 2)];
      c = wmma4(a, b, c);
    }
    const float bv = bh1[n0 + lo];
#pragma unroll
    for (int r = 0; r < 8; ++r)
      lds_h[r + 8 * hi][n0 + lo] = fmaxf(c[r] + bv, 0.0f);
  }
  __syncthreads();

  // ---- stage 3: logits = h @ Wh2 + bh2 (16 rows, 32-length dot each)
  if (wave == 0 && lane < 16) {
    float acc = bh2[0];
#pragma unroll
    for (int k = 0; k < HID / 2; ++k) acc += lds_h[lane][k] * Wh2[k];
    out[m0 + lane] = acc;
  }
}

// ---------------------------------------------------------------------------
extern "C" void kernel_launch(void* const* d_in, const int* in_sizes, int n_in,
                              void* d_out, int out_size, void* d_ws, size_t ws_size,
                              hipStream_t stream) {
  const float* loc_feat = (const float*)d_in[0];
  const float* evt_feat = (const float*)d_in[1];
  const int*   qid      = (const int*)d_in[2];
  const int*   edge     = (const int*)d_in[3];
  const float* emb      = (const float*)d_in[4];
  const float* W_loc    = (const float*)d_in[5];
  const float* b_loc    = (const float*)d_in[6];
  const float* W_evt    = (const float*)d_in[7];
  const float* b_evt    = (const float*)d_in[8];
  const float* W_l      = (const float*)d_in[9];
  const float* b_l      = (const float*)d_in[10];
  const float* W_r      = (const float*)d_in[11];
  const float* W_h1     = (const float*)d_in[12];
  const float* b_h1     = (const float*)d_in[13];
  const float* W_h2     = (const float*)d_in[14];
  const float* b_h2     = (const float*)d_in[15];
  float* out = (float*)d_out;

  // workspace layout (fp32): evt_x | loc_x | agg | cnt  (~77 MB total)
  float* evt_x = (float*)d_ws;
  float* loc_x = evt_x + (size_t)N_EVT * HID;
  float* agg   = loc_x + (size_t)N_LOC * HID;
  float* cnt   = agg   + (size_t)N_LOC * HID;

  const long long nz = (long long)N_LOC * HID + N_LOC;  // agg + cnt contiguous
  zero_f32<<<2048, 256, 0, stream>>>(agg, nz);

  evt_gemm<<<N_EVT / 16, 128, 0, stream>>>(evt_feat, W_evt, b_evt, evt_x);
  loc_gemm<<<N_LOC / 16, 128, 0, stream>>>(loc_feat, qid, emb, W_loc, b_loc, loc_x);

  scatter_edges<<<(N_EDGE * 16) / 256, 256, 0, stream>>>(edge, evt_x, agg, cnt);

  sage_head<<<N_LOC / 16, 128, 0, stream>>>(agg, cnt, loc_x, W_l, b_l, W_r,
                                            W_h1, b_h1, W_h2, b_h2, out);
}